// LacssModel_47966194762317
// MI455X (gfx1250) — compile-verified
//
#include <hip/hip_runtime.h>

typedef __attribute__((ext_vector_type(2))) float v2f;
typedef __attribute__((ext_vector_type(8))) float v8f;

#define B_  64
#define N_  1024
#define M_  2048
#define THR2 256.0f   // 16.0^2

// ds_swizzle_b32 group-of-32 xor patterns: imm = (xor_mask<<10) | and_mask(0x1f)
#define SWZ_X1 0x041F
#define SWZ_X2 0x081F
#define SWZ_X4 0x101F
#define SWZ_X8 0x201F

// ---------------------------------------------------------------------------
// Kernel 1: reset per-gt "lowest matched pred index" table to sentinel M.
// ---------------------------------------------------------------------------
__global__ __launch_bounds__(256) void lacss_init_ws(int* __restrict__ ws) {
  int t = blockIdx.x * 256 + threadIdx.x;
  if (t < B_ * N_) ws[t] = M_;
}

// ---------------------------------------------------------------------------
// Kernel 2: WMMA distance tiles + per-pred argmin + per-gt atomicMin.
// grid = B * 16 blocks, 256 threads (8 waves). Wave w handles pred tile
// (group*8 + w): 16 preds vs all 1024 gts = 64 WMMA ops.
//
// d2 tile via V_WMMA_F32_16X16X4_F32 with
//   A row m = ( px, py, |p|^2, 1 )          (K=4, zero-padded trick folded)
//   B col n = ( -2gx, -2gy, 1, |g|^2 )
// A layout (ISA 7.12.2): lanes 0-15 hold K=0,1 ; lanes 16-31 hold K=2,3.
// B mirrors A; D vgpr r: lanes 0-15 -> M=r, lanes 16-31 -> M=r+8.
// ---------------------------------------------------------------------------
__global__ __launch_bounds__(256) void lacss_match_wmma(
    const float2* __restrict__ gt,     // [B,N]
    const float2* __restrict__ pred,   // [B,M]
    int* __restrict__ ws) {            // [B,N] min matched pred idx
  // Two pre-swizzled B-operand tables; sB1 shifted by 32 banks (128 B) so
  // lanes 0-15 (reading sB0) and lanes 16-31 (reading sB1) cover disjoint
  // bank halves -> conflict-free ds_load_b64.
  __shared__ float sRaw[2 * N_ + 32 + 2 * N_];
  float2* sB0 = (float2*)sRaw;                   // (-2gx, -2gy)
  float2* sB1 = (float2*)(sRaw + 2 * N_ + 32);   // (1, |g|^2)

  const int b   = blockIdx.x >> 4;
  const int grp = blockIdx.x & 15;

  const float2* gtb = gt + b * N_;
  for (int i = threadIdx.x; i < N_; i += 256) {
    float2 g = gtb[i];
    sB0[i] = make_float2(-2.0f * g.x, -2.0f * g.y);
    sB1[i] = make_float2(1.0f, g.x * g.x + g.y * g.y);
  }
  __syncthreads();

  const int lane  = threadIdx.x & 31;
  const int half  = lane >> 4;          // 0: K=0,1 ; 1: K=2,3
  const int lm    = lane & 15;
  const int ptile = grp * 8 + (threadIdx.x >> 5);   // 0..127
  const int prow  = ptile * 16 + lm;

  float2 p = pred[(size_t)b * M_ + prow];
  v2f a;
  a.x = half ? (p.x * p.x + p.y * p.y) : p.x;
  a.y = half ? 1.0f                    : p.y;

  const float2* myB = half ? sB1 : sB0;  // per-lane base picked once

  float bestv[8];
  int   besti[8];
#pragma unroll
  for (int r = 0; r < 8; ++r) { bestv[r] = 3.4e38f; besti[r] = 0; }

  // --- peeled t = 0 (reuse_a must be 0: previous instr is not this WMMA) ---
  {
    float2 g2 = myB[lm];
    v2f bb; bb.x = g2.x; bb.y = g2.y;
    v8f c = {};
    v8f d = __builtin_amdgcn_wmma_f32_16x16x4_f32(
        false, a, false, bb, (short)0, c, /*reuse_a=*/false, false);
#pragma unroll
    for (int r = 0; r < 8; ++r) {
      float v = d[r];
      if (v < bestv[r]) { bestv[r] = v; besti[r] = lm; }
    }
  }
  // --- steady state: identical WMMA each iter, A operand loop-invariant ----
  for (int t = 1; t < N_ / 16; ++t) {
    float2 g2 = myB[t * 16 + lm];       // ds_load_b64, bank-conflict free
    v2f bb; bb.x = g2.x; bb.y = g2.y;
    v8f c = {};
    v8f d = __builtin_amdgcn_wmma_f32_16x16x4_f32(
        false, a, false, bb, (short)0, c, /*reuse_a=*/true, false);
    const int n = t * 16 + lm;
#pragma unroll
    for (int r = 0; r < 8; ++r) {
      float v = d[r];
      if (v < bestv[r]) { bestv[r] = v; besti[r] = n; }  // strict < => first min
    }
  }

  // Pack (distance bits, N index) into u64 keys: distances are >= 0 so the
  // fp32 bit pattern orders correctly as u32; u64-min gives (min d2, lowest N)
  // branch-free, matching jnp.argmin tie-breaking.
  unsigned long long key[8];
#pragma unroll
  for (int r = 0; r < 8; ++r)
    key[r] = ((unsigned long long)__float_as_uint(bestv[r]) << 32) |
             (unsigned)besti[r];

  // xor-swizzle reduction across the 16 lanes of each half (masks 1,2,4,8
  // never cross the half boundary in wave32).
#define REDUCE_STEP(IMM)                                                     \
  _Pragma("unroll")                                                          \
  for (int r = 0; r < 8; ++r) {                                              \
    unsigned lo = (unsigned)__builtin_amdgcn_ds_swizzle(                     \
        (int)(unsigned)key[r], (IMM));                                       \
    unsigned hi = (unsigned)__builtin_amdgcn_ds_swizzle(                     \
        (int)(unsigned)(key[r] >> 32), (IMM));                               \
    unsigned long long o = ((unsigned long long)hi << 32) | lo;              \
    key[r] = (o < key[r]) ? o : key[r];                                      \
  }
  REDUCE_STEP(SWZ_X1)
  REDUCE_STEP(SWZ_X2)
  REDUCE_STEP(SWZ_X4)
  REDUCE_STEP(SWZ_X8)
#undef REDUCE_STEP

  // One lane per half publishes its 8 pred rows (D rows M=r / M=r+8).
  if (lm == 0) {
#pragma unroll
    for (int r = 0; r < 8; ++r) {
      float v  = __uint_as_float((unsigned)(key[r] >> 32));
      int   gi = (int)(unsigned)key[r];
      int   j  = ptile * 16 + r + 8 * half;   // global pred index
      if (v < THR2) atomicMin(&ws[b * N_ + gi], j);
    }
  }
}

// ---------------------------------------------------------------------------
// Kernel 3: out = (idx < M) ? pred[idx] : gt
// ---------------------------------------------------------------------------
__global__ __launch_bounds__(256) void lacss_finalize(
    const float2* __restrict__ gt, const float2* __restrict__ pred,
    const int* __restrict__ ws, float2* __restrict__ out) {
  int t = blockIdx.x * 256 + threadIdx.x;
  if (t >= B_ * N_) return;
  int b   = t >> 10;                        // t / N_
  int idx = ws[t];
  out[t] = (idx < M_) ? pred[(size_t)b * M_ + idx] : gt[t];
}

extern "C" void kernel_launch(void* const* d_in, const int* in_sizes, int n_in,
                              void* d_out, int out_size, void* d_ws, size_t ws_size,
                              hipStream_t stream) {
  const float2* gt   = (const float2*)d_in[0];   // [64,1024,2] f32
  const float2* pred = (const float2*)d_in[1];   // [64,2048,2] f32
  float2* out = (float2*)d_out;                  // [64,1024,2] f32
  int* ws = (int*)d_ws;                          // B*N ints = 256 KB

  lacss_init_ws<<<(B_ * N_ + 255) / 256, 256, 0, stream>>>(ws);
  lacss_match_wmma<<<B_ * 16, 256, 0, stream>>>(gt, pred, ws);
  lacss_finalize<<<(B_ * N_ + 255) / 256, 256, 0, stream>>>(gt, pred, ws, out);
}